// GCNEncoder_47785806135533
// MI455X (gfx1250) — compile-verified
//
#include <hip/hip_runtime.h>
#include <hip/hip_bf16.h>
#include <stdint.h>

#define D 128
#define WPAD 132
#define BN_EPS 1e-5f

typedef __attribute__((ext_vector_type(2))) float v2f;
typedef __attribute__((ext_vector_type(8))) float v8f;

// ---------------- utility kernels ----------------
__global__ void fill_kernel(float* __restrict__ p, float v, long n) {
    long i = (long)blockIdx.x * blockDim.x + threadIdx.x;
    if (i < n) p[i] = v;
}

__global__ void degree_kernel(const int* __restrict__ dst, float* __restrict__ deg, int E) {
    int i = blockIdx.x * blockDim.x + threadIdx.x;
    if (i < E) unsafeAtomicAdd(&deg[dst[i]], 1.0f);
}

__global__ void rsqrt_kernel(float* __restrict__ p, int n) {
    int i = blockIdx.x * blockDim.x + threadIdx.x;
    if (i < n) p[i] = rsqrtf(p[i]);
}

// ---------------- GEMM + self-term fusion ----------------
// O[N,128]  = A[N,128] @ W[128,128]                  (needed by edge scatter)
// AG[N,128] = O * dinv^2 + bias                       (agg initialized with self-loop + bias)
// Block = 8 waves (wave32), 32 rows (two 16-row tiles). Wave w owns cols [16w,16w+16).
// B operand (W fragments) register-resident for whole K; A tile staged to LDS with
// gfx1250 async-load-to-LDS; K consumed by V_WMMA_F32_16X16X4_F32 (full fp32).
__global__ __launch_bounds__(256) void gemm128_wmma_f32(
    const float* __restrict__ A, const float* __restrict__ W,
    float* __restrict__ O, float* __restrict__ AG,
    const float* __restrict__ dinv, const float* __restrict__ bias, int nrows) {
    __shared__ float sA[32 * WPAD];   // 32x128 A tile, padded (16.9 KB)

    const int t = threadIdx.x;
    const int row0 = blockIdx.x * 32;

    // stage A tile: 4096 floats, 16 per thread, via async global->LDS (ASYNCcnt path)
#pragma unroll
    for (int i = 0; i < 16; ++i) {
        int idx = t + i * 256;
        int r = idx >> 7, c = idx & 127;
        int gr = row0 + r;
        if (gr >= nrows) gr = nrows - 1;          // clamp; extra rows never stored
        unsigned ldsoff = (unsigned)(uintptr_t)(&sA[r * WPAD + c]);
        const float* gp = A + (long)gr * D + c;
        asm volatile("global_load_async_to_lds_b32 %0, %1, off"
                     :: "v"(ldsoff), "v"(gp) : "memory");
    }

    const int lane = t & 31;
    const int wave = t >> 5;
    const int m    = lane & 15;   // A row within tile / B col within wave tile
    const int half = lane >> 4;   // K sub-pair selector (0 or 1)
    const int col  = wave * 16 + m;

    // preload all B fragments for this wave's 16 columns (64 floats/lane, L0/L2-served)
    v2f bfrag[32];
#pragma unroll
    for (int j = 0; j < 32; ++j) {
        int k = 4 * j + 2 * half;
        bfrag[j].x = W[(long)k * D + col];
        bfrag[j].y = W[(long)(k + 1) * D + col];
    }

    asm volatile("s_wait_asynccnt 0x0" ::: "memory");
    __syncthreads();

    v8f acc0 = {};
    v8f acc1 = {};
#pragma unroll
    for (int j = 0; j < 32; ++j) {
        int k = 4 * j + 2 * half;
        v2f a0, a1;
        a0.x = sA[m * WPAD + k];
        a0.y = sA[m * WPAD + k + 1];
        a1.x = sA[(16 + m) * WPAD + k];
        a1.y = sA[(16 + m) * WPAD + k + 1];
        acc0 = __builtin_amdgcn_wmma_f32_16x16x4_f32(false, a0, false, bfrag[j],
                                                     (short)0, acc0, false, false);
        acc1 = __builtin_amdgcn_wmma_f32_16x16x4_f32(false, a1, false, bfrag[j],
                                                     (short)0, acc1, false, false);
    }

    // C/D layout: VGPR i -> row (i + 8*half), col = lane&15 within wave tile
    const float bcol = bias[col];
#pragma unroll
    for (int i = 0; i < 8; ++i) {
        int gr = row0 + half * 8 + i;
        if (gr < nrows) {
            float o = acc0[i];
            float dv = dinv[gr];
            O[(long)gr * D + col]  = o;
            AG[(long)gr * D + col] = o * dv * dv + bcol;
        }
        int gr1 = row0 + 16 + half * 8 + i;
        if (gr1 < nrows) {
            float o = acc1[i];
            float dv = dinv[gr1];
            O[(long)gr1 * D + col]  = o;
            AG[(long)gr1 * D + col] = o * dv * dv + bcol;
        }
    }
}

// ---------------- edge scatter: agg[dst] += H[src] * dinv[src]*dinv[dst] ----------------
__global__ void scatter_kernel(const float* __restrict__ H,
                               const int* __restrict__ src, const int* __restrict__ dst,
                               const float* __restrict__ dinv, float* __restrict__ agg,
                               long total) {
    long gid = (long)blockIdx.x * blockDim.x + threadIdx.x;
    if (gid >= total) return;
    int e = (int)(gid >> 7);
    int c = (int)(gid & 127);
    int s = src[e], d = dst[e];
    float w = dinv[s] * dinv[d];
    unsafeAtomicAdd(&agg[(long)d * D + c], H[(long)s * D + c] * w);
}

// ---------------- BN(eval) + ReLU: hbuf = relu(gamma*(agg-mean)*rsqrt(var+eps)+beta) ----
__global__ void bn_relu_kernel(const float* __restrict__ agg,
                               const float* __restrict__ gamma,
                               const float* __restrict__ beta,
                               const float* __restrict__ mean,
                               const float* __restrict__ var,
                               float* __restrict__ out, long total) {
    long gid = (long)blockIdx.x * blockDim.x + threadIdx.x;
    if (gid >= total) return;
    int c = (int)(gid & 127);
    float v = agg[gid];
    v = gamma[c] * (v - mean[c]) * rsqrtf(var[c] + BN_EPS) + beta[c];
    out[gid] = fmaxf(v, 0.0f);
}

// ---------------- pooling ----------------
__global__ void pool_sum_kernel(const float* __restrict__ H, const int* __restrict__ batch,
                                float* __restrict__ sums, long total) {
    long gid = (long)blockIdx.x * blockDim.x + threadIdx.x;
    if (gid >= total) return;
    int i = (int)(gid >> 7), c = (int)(gid & 127);
    unsafeAtomicAdd(&sums[(long)batch[i] * D + c], H[gid]);
}

__global__ void pool_cnt_kernel(const int* __restrict__ batch, float* __restrict__ cnt, int n) {
    int i = blockIdx.x * blockDim.x + threadIdx.x;
    if (i < n) unsafeAtomicAdd(&cnt[batch[i]], 1.0f);
}

__global__ void pool_div_kernel(float* __restrict__ out, const float* __restrict__ cnt, long total) {
    long gid = (long)blockIdx.x * blockDim.x + threadIdx.x;
    if (gid >= total) return;
    out[gid] = out[gid] / fmaxf(cnt[gid >> 7], 1.0f);
}

// ---------------- launcher ----------------
extern "C" void kernel_launch(void* const* d_in, const int* in_sizes, int n_in,
                              void* d_out, int out_size, void* d_ws, size_t ws_size,
                              hipStream_t stream) {
    const float* x     = (const float*)d_in[0];
    const int*   ei    = (const int*)d_in[1];
    const int*   batch = (const int*)d_in[2];
    const float* Ws    = (const float*)d_in[3];
    const float* bs    = (const float*)d_in[4];
    const float* gam   = (const float*)d_in[5];
    const float* bet   = (const float*)d_in[6];
    const float* mea   = (const float*)d_in[7];
    const float* var   = (const float*)d_in[8];

    const int N = in_sizes[0] / D;
    const int E = in_sizes[1] / 2;
    const int L = in_sizes[3] / (D * D);
    const int G = out_size / D;

    const int* src = ei;
    const int* dst = ei + E;

    float* ws   = (float*)d_ws;
    float* hbuf = ws;                         // N*D
    float* hmul = hbuf + (size_t)N * D;       // N*D
    float* agg  = hmul + (size_t)N * D;       // N*D
    float* dinv = agg  + (size_t)N * D;       // N
    float* cnt  = dinv + N;                   // G

    float* out = (float*)d_out;

    const long ND = (long)N * D;
    const long ED = (long)E * D;
    const long GD = (long)G * D;
    const int  T  = 256;
    auto blks = [](long n, int t) { return (unsigned)((n + t - 1) / t); };

    // degrees (self-loop baked in as 1.0) -> dinv
    fill_kernel<<<blks(N, T), T, 0, stream>>>(dinv, 1.0f, N);
    degree_kernel<<<blks(E, T), T, 0, stream>>>(dst, dinv, E);
    rsqrt_kernel<<<blks(N, T), T, 0, stream>>>(dinv, N);

    // zero pooling accumulators (d_out is fully rewritten each call)
    fill_kernel<<<blks(GD, T), T, 0, stream>>>(out, 0.0f, GD);
    fill_kernel<<<blks(G, T), T, 0, stream>>>(cnt, 0.0f, G);

    const unsigned gemm_blocks = (unsigned)((N + 31) / 32);
    for (int l = 0; l < L; ++l) {
        const float* hin = (l == 0) ? x : hbuf;
        // hmul = hin @ W_l ; agg = hmul*dinv^2 + b_l (self-loop term pre-seeded)
        gemm128_wmma_f32<<<gemm_blocks, 256, 0, stream>>>(
            hin, Ws + (size_t)l * D * D, hmul, agg, dinv, bs + (size_t)l * D, N);
        scatter_kernel<<<blks(ED, T), T, 0, stream>>>(hmul, src, dst, dinv, agg, ED);
        if (l < L - 1) {
            bn_relu_kernel<<<blks(ND, T), T, 0, stream>>>(
                agg, gam + (size_t)l * D, bet + (size_t)l * D,
                mea + (size_t)l * D, var + (size_t)l * D, hbuf, ND);
        }
    }

    // global mean pool over final conv output (in agg)
    pool_sum_kernel<<<blks(ND, T), T, 0, stream>>>(agg, batch, out, ND);
    pool_cnt_kernel<<<blks(N, T), T, 0, stream>>>(batch, cnt, N);
    pool_div_kernel<<<blks(GD, T), T, 0, stream>>>(out, cnt, GD);
}